// Gate_60421599920823
// MI455X (gfx1250) — compile-verified
//
#include <hip/hip_runtime.h>

typedef __attribute__((ext_vector_type(2))) float v2f;
typedef __attribute__((ext_vector_type(8))) float v8f;

#define N_TOK 8192
#define DIM   4096
#define NEXP  64
#define SCALE 1.0f

// One workgroup = 16 token rows, 2 waves (wave32), split over K (not experts):
//   wave w accumulates logits for ALL 64 experts over K in [w*2048, (w+1)*2048)
//   -> per k-step: 1x A load + 4x B load + 4x V_WMMA_F32_16X16X4_F32
//   -> 1.25 global_load_b64 per wmma, and no duplicate A loads in the WG.
// Wave 1 spills partial accumulators to LDS; wave 0 reduces, applies sigmoid,
// and does the biased top-2 + normalization.
__global__ __launch_bounds__(64) void gate_wmma_kernel(
    const float* __restrict__ x,       // (N, D) fp32
    const float* __restrict__ w,       // (E, D) fp32
    const float* __restrict__ bias,    // (E)    fp32
    float* __restrict__ out_weights,   // (N, 2) fp32
    int*   __restrict__ out_indices,   // (N, 2) int32 bits
    float* __restrict__ out_scores)    // (N, E) fp32
{
    __shared__ float lds_buf[16][NEXP + 1];      // partials, then scores; +1 pad

    const int lane  = threadIdx.x & 31;
    const int wave  = threadIdx.x >> 5;          // 0..1
    const int m0    = blockIdx.x * 16;           // token-row tile base
    const int kbase = wave * (DIM / 2);          // K half owned by this wave

    const int r     = lane & 15;                 // A: row M; B: expert col
    const int khalf = (lane >> 4) * 2;           // K sub-offset within step: 0 or 2

    const float* xp  = x + (size_t)(m0 + r) * DIM + kbase + khalf;
    const float* wp0 = w + (size_t)( 0 + r) * DIM + kbase + khalf;  // experts  0..15
    const float* wp1 = w + (size_t)(16 + r) * DIM + kbase + khalf;  // experts 16..31
    const float* wp2 = w + (size_t)(32 + r) * DIM + kbase + khalf;  // experts 32..47
    const float* wp3 = w + (size_t)(48 + r) * DIM + kbase + khalf;  // experts 48..63

    v8f c0 = {}, c1 = {}, c2 = {}, c3 = {};
    #pragma unroll 4
    for (int k = 0; k < DIM / 2; k += 4) {
        if ((k & 15) == 0) {
            // stream-prefetch x ~4KB ahead (speculative: tail overrun is dropped)
            __builtin_prefetch(xp + k + 1024, 0, 0);
        }
        v2f a  = *(const v2f*)(xp  + k);         // A fragment: x[m, k..k+3]
        v2f b0 = *(const v2f*)(wp0 + k);         // B fragments (= W^T columns)
        v2f b1 = *(const v2f*)(wp1 + k);
        v2f b2 = *(const v2f*)(wp2 + k);
        v2f b3 = *(const v2f*)(wp3 + k);
        // 8 args: (neg_a, A, neg_b, B, c_mod, C, reuse_a, reuse_b)
        c0 = __builtin_amdgcn_wmma_f32_16x16x4_f32(false, a, false, b0, (short)0, c0, false, false);
        c1 = __builtin_amdgcn_wmma_f32_16x16x4_f32(false, a, false, b1, (short)0, c1, false, false);
        c2 = __builtin_amdgcn_wmma_f32_16x16x4_f32(false, a, false, b2, (short)0, c2, false, false);
        c3 = __builtin_amdgcn_wmma_f32_16x16x4_f32(false, a, false, b3, (short)0, c3, false, false);
    }

    // C layout: VGPR j, lane l -> row (j + 8*(l>=16)), col (tile*16 + (l&15))
    const int mhi = (lane >> 4) * 8;
    const int col = lane & 15;

    if (wave == 1) {                             // spill high-K partials
        #pragma unroll
        for (int j = 0; j < 8; ++j) {
            const int m = mhi + j;
            lds_buf[m][col +  0] = c0[j];
            lds_buf[m][col + 16] = c1[j];
            lds_buf[m][col + 32] = c2[j];
            lds_buf[m][col + 48] = c3[j];
        }
    }
    __syncthreads();

    if (wave == 0) {                             // reduce + sigmoid + scores
        #pragma unroll
        for (int j = 0; j < 8; ++j) {
            const int m = mhi + j;
            float l0 = c0[j] + lds_buf[m][col +  0];
            float l1 = c1[j] + lds_buf[m][col + 16];
            float l2 = c2[j] + lds_buf[m][col + 32];
            float l3 = c3[j] + lds_buf[m][col + 48];
            float s0 = 1.0f / (1.0f + __expf(-l0));
            float s1 = 1.0f / (1.0f + __expf(-l1));
            float s2 = 1.0f / (1.0f + __expf(-l2));
            float s3 = 1.0f / (1.0f + __expf(-l3));
            lds_buf[m][col +  0] = s0;           // reads above precede these writes
            lds_buf[m][col + 16] = s1;
            lds_buf[m][col + 32] = s2;
            lds_buf[m][col + 48] = s3;
            float* srow = out_scores + (size_t)(m0 + m) * NEXP;
            srow[col +  0] = s0;
            srow[col + 16] = s1;
            srow[col + 32] = s2;
            srow[col + 48] = s3;
        }
    }
    __syncthreads();

    // Top-2 (on score + bias), gather unbiased scores, normalize.
    // Ties break toward the lower index (strict '>'), matching jax.lax.top_k.
    if (wave == 0 && lane < 16) {
        const int m = lane;
        float best1 = -1e30f, best2 = -1e30f;
        int   i1 = 0, i2 = 0;
        #pragma unroll 4
        for (int e = 0; e < NEXP; ++e) {
            float bv = lds_buf[m][e] + bias[e];  // bias[e] lane-uniform -> s_load
            if (bv > best1) { best2 = best1; i2 = i1; best1 = bv; i1 = e; }
            else if (bv > best2) { best2 = bv; i2 = e; }
        }
        float w1 = lds_buf[m][i1];
        float w2 = lds_buf[m][i2];
        float inv = SCALE / (w1 + w2);
        const int row = m0 + m;
        out_weights[row * 2 + 0] = w1 * inv;
        out_weights[row * 2 + 1] = w2 * inv;
        out_indices[row * 2 + 0] = i1;
        out_indices[row * 2 + 1] = i2;
    }
}

extern "C" void kernel_launch(void* const* d_in, const int* in_sizes, int n_in,
                              void* d_out, int out_size, void* d_ws, size_t ws_size,
                              hipStream_t stream) {
    const float* x    = (const float*)d_in[0];   // (8192, 4096)
    const float* w    = (const float*)d_in[1];   // (64, 4096)
    const float* bias = (const float*)d_in[2];   // (64)

    // d_out = [weights (N*2 f32) | indices (N*2 int32 bits) | scores (N*E f32)]
    float* base        = (float*)d_out;
    float* out_weights = base;
    int*   out_indices = (int*)(base + (size_t)N_TOK * 2);
    float* out_scores  = base + (size_t)N_TOK * 4;

    dim3 grid(N_TOK / 16);   // 512 workgroups
    dim3 block(64);          // 2 waves of 32
    gate_wmma_kernel<<<grid, block, 0, stream>>>(x, w, bias,
                                                 out_weights, out_indices, out_scores);
}